// GATConv_6390911336706
// MI455X (gfx1250) — compile-verified
//
#include <hip/hip_runtime.h>
#include <hip/hip_bf16.h>

// GAT layer for MI455X (gfx1250, wave32).
// Phase 1: h = x @ Wflat  (50000x512x256 fp32 GEMM) via V_WMMA_F32_16X16X4_F32,
//          with GLOBAL_LOAD_ASYNC_TO_LDS_B128 staging (ASYNCcnt-tracked).
// Phase 2: per-(node,head) attention logits s_src/s_dst.
// Phase 3: per-node 32-wide softmax + neighbor aggregation (L2-resident gather).

#define NN      50000
#define FIN     512
#define FOUT    256
#define NHEAD   8
#define DHD     32
#define DEG     32
#define BM      128
#define BK      16

typedef __attribute__((ext_vector_type(2))) float v2f;
typedef __attribute__((ext_vector_type(8))) float v8f;

// The async-LDS builtin's params are pointers to GCC-style 16B int vectors:
//   (v4i addrspace(1)*, v4i addrspace(3)*, imm offset, imm cpol)
typedef int v4i_vs __attribute__((vector_size(16)));
typedef __attribute__((address_space(1))) v4i_vs* g_v4i_ptr;
typedef __attribute__((address_space(3))) v4i_vs* l_v4i_ptr;
typedef __attribute__((address_space(1))) void* g_void_ptr;
typedef __attribute__((address_space(3))) void* l_void_ptr;

#if __has_builtin(__builtin_amdgcn_global_load_async_to_lds_b128) && \
    __has_builtin(__builtin_amdgcn_s_wait_asynccnt)
#define USE_ASYNC_LDS 1
#endif

// 16-byte global -> LDS copy. Async path writes LDS directly (no VGPR round
// trip, no per-load s_wait_loadcnt); tracked with ASYNCcnt.
__device__ __forceinline__ void copy16_g2lds(const float* g, float* l) {
#ifdef USE_ASYNC_LDS
    g_v4i_ptr gv = (g_v4i_ptr)(g_void_ptr)g;
    l_v4i_ptr lv = (l_v4i_ptr)(l_void_ptr)l;
    __builtin_amdgcn_global_load_async_to_lds_b128(gv, lv, /*offset=*/0, /*cpol=*/0);
#else
    float4 v = *(const float4*)g;
    l[0] = v.x; l[1] = v.y; l[2] = v.z; l[3] = v.w;
#endif
}

__device__ __forceinline__ void staging_fence() {
#ifdef USE_ASYNC_LDS
    __builtin_amdgcn_s_wait_asynccnt(0);
#endif
    __syncthreads();
}

// ---------------------------------------------------------------------------
// Kernel 1: tiled fp32 WMMA GEMM.  h[n][c] = sum_f x[n][f] * Wflat[f][c],
// where Wflat[f][hh*32+d] = W[hh][f][d]  (W laid out (H, F_IN, DH)).
// Block: 256 threads = 8 waves.  Block tile 128 rows x 256 cols.
// Wave w owns rows [16w,16w+16) of the block and all 16 column tiles
// (16 accumulators of 16x16 f32 = 128 VGPRs).
// ---------------------------------------------------------------------------
__global__ __launch_bounds__(256) void gat_gemm(const float* __restrict__ x,
                                                const float* __restrict__ W,
                                                float* __restrict__ h) {
    __shared__ float Alds[BM][BK + 1];     // pad 16->17: conflict-free 16-lane stride
    __shared__ float Blds[BK][FOUT + 16];  // pad 256->272: khalf groups hit disjoint banks

    const int tid   = threadIdx.x;
    const int wave  = tid >> 5;
    const int lane  = tid & 31;
    const int m     = lane & 15;   // M (for A) / N (for B) index of this lane
    const int khalf = lane >> 4;   // 0: K{0,1}, 1: K{2,3} within a 4-wide K chunk
    const int row0  = blockIdx.x * BM;

    v8f acc[16];
#pragma unroll
    for (int t = 0; t < 16; ++t)
#pragma unroll
        for (int r = 0; r < 8; ++r) acc[t][r] = 0.0f;

    for (int kk = 0; kk < FIN; kk += BK) {
        // --- stage A: 128x16 fp32 (512 float4 slots, 2 per thread) ---
        // Out-of-range rows are clamped (not zeroed): row r of A only feeds
        // output row row0+r, which the store guard discards anyway.
#pragma unroll
        for (int i = 0; i < 2; ++i) {
            int f4 = tid + i * 256;
            int r  = f4 >> 2;          // 0..127
            int kc = (f4 & 3) * 4;     // 0,4,8,12
            int grow = row0 + r;
            if (grow >= NN) grow = NN - 1;
            copy16_g2lds(x + (size_t)grow * FIN + kk + kc, &Alds[r][kc]);
        }
        // --- stage B: 16x256 fp32 (1024 float4 slots, 4 per thread) ---
        // Bflat[f][c] = W[(c>>5)][f][c&31]; 4 consecutive c stay inside one head.
#pragma unroll
        for (int i = 0; i < 4; ++i) {
            int f4 = tid + i * 256;
            int kr = f4 >> 6;          // 0..15
            int c  = (f4 & 63) * 4;    // 0..252 (multiple of 4)
            const float* src = W + (size_t)(c >> 5) * (FIN * DHD)
                                 + (size_t)(kk + kr) * DHD + (c & 31);
            copy16_g2lds(src, &Blds[kr][c]);
        }
        staging_fence();

        // --- compute: 4 K-chunks of 4, 16 col tiles each => 64 WMMAs / wave ---
#pragma unroll
        for (int kc = 0; kc < BK; kc += 4) {
            const int ka = kc + 2 * khalf;
            v2f a;
            a.x = Alds[16 * wave + m][ka];
            a.y = Alds[16 * wave + m][ka + 1];
#pragma unroll
            for (int t = 0; t < 16; ++t) {
                v2f b;
                b.x = Blds[ka][16 * t + m];
                b.y = Blds[ka + 1][16 * t + m];
                // 8 args: (neg_a, A, neg_b, B, c_mod, C, reuse_a, reuse_b)
                acc[t] = __builtin_amdgcn_wmma_f32_16x16x4_f32(
                    false, a, false, b, (short)0, acc[t], false, false);
            }
        }
        __syncthreads();
    }

    // --- store: C/D layout: VGPR r holds row (r + 8*khalf), col = 16t + m ---
    const int rbase = row0 + 16 * wave + 8 * khalf;
#pragma unroll
    for (int t = 0; t < 16; ++t)
#pragma unroll
        for (int r = 0; r < 8; ++r) {
            int grow = rbase + r;
            if (grow < NN) h[(size_t)grow * FOUT + 16 * t + m] = acc[t][r];
        }
}

// ---------------------------------------------------------------------------
// Kernel 2: attention logits.  s_src[n*8+hh] = h[n][hh*32:..] . a[hh][0:32]
//                              s_dst[n*8+hh] = h[n][hh*32:..] . a[hh][32:64]
// ---------------------------------------------------------------------------
__global__ void gat_scores(const float* __restrict__ h,
                           const float* __restrict__ a,
                           float* __restrict__ ssrc,
                           float* __restrict__ sdst) {
    int g = blockIdx.x * blockDim.x + threadIdx.x;
    if (g >= NN * NHEAD) return;
    int n  = g >> 3;
    int hh = g & 7;
    const float* hp   = h + (size_t)n * FOUT + hh * DHD;
    const float* asrc = a + hh * (2 * DHD);
    const float* adst = asrc + DHD;
    float s0 = 0.f, s1 = 0.f;
#pragma unroll
    for (int d = 0; d < DHD; ++d) {
        float v = hp[d];
        s0 += v * asrc[d];
        s1 += v * adst[d];
    }
    ssrc[g] = s0;
    sdst[g] = s1;
}

// ---------------------------------------------------------------------------
// Kernel 3: per-node softmax over its 32 contiguous edges + aggregation.
// One wave per node (8 nodes / 256-thread block).  Edges of node n are
// e = n*32+j, j=0..31 (row = repeat(arange(N),32)); col gives the neighbor.
// score[j][hh] = leaky_relu(s_src[n,hh] + s_dst[col_j,hh]); softmax over j;
// out[n][hh*32+d] = sum_j alpha[j][hh] * h[col_j][hh*32+d].
// Column c of out has head c>>5 since DH == 32.
// ---------------------------------------------------------------------------
__global__ __launch_bounds__(256) void gat_aggregate(const float* __restrict__ h,
                                                     const float* __restrict__ ssrc,
                                                     const float* __restrict__ sdst,
                                                     const int*   __restrict__ col,
                                                     float* __restrict__ out) {
    __shared__ float sc[8][DEG][NHEAD];   // scores -> alphas
    __shared__ int   cidx[8][DEG];
    __shared__ float mx[8][NHEAD];
    __shared__ float rden[8][NHEAD];

    const int wave = threadIdx.x >> 5;
    const int lane = threadIdx.x & 31;
    const int n    = blockIdx.x * 8 + wave;   // 6250 * 8 == 50000 exactly

    // 1) gather neighbor ids + raw scores
    const int cj = col[(size_t)n * DEG + lane];
    cidx[wave][lane] = cj;
    const float* ss = ssrc + (size_t)n * NHEAD;
    const float* sd = sdst + (size_t)cj * NHEAD;
#pragma unroll
    for (int hh = 0; hh < NHEAD; ++hh) {
        float e = ss[hh] + sd[hh];
        sc[wave][lane][hh] = (e > 0.f) ? e : 0.01f * e;   // leaky relu
    }
    __syncthreads();

    // 2) per-head max + sum(exp) over the 32 edges (lanes 0..7, one head each)
    if (lane < NHEAD) {
        float m_ = -1e30f;
        for (int j = 0; j < DEG; ++j) m_ = fmaxf(m_, sc[wave][j][lane]);
        float s_ = 0.f;
        for (int j = 0; j < DEG; ++j) s_ += __expf(sc[wave][j][lane] - m_);
        mx[wave][lane]   = m_;
        rden[wave][lane] = 1.f / s_;
    }
    __syncthreads();

    // 3) alphas in place
#pragma unroll
    for (int hh = 0; hh < NHEAD; ++hh)
        sc[wave][lane][hh] = __expf(sc[wave][lane][hh] - mx[wave][hh]) * rden[wave][hh];
    __syncthreads();

    // 4) aggregate: lane i owns columns {32t + i}; gathers are 128B coalesced,
    //    all h traffic is L2-resident (51.2 MB << 192 MB L2).
    float acc[NHEAD];
#pragma unroll
    for (int t = 0; t < NHEAD; ++t) acc[t] = 0.f;
    for (int jj = 0; jj < DEG; ++jj) {
        const int c = cidx[wave][jj];
        const float* hp = h + (size_t)c * FOUT + lane;
#pragma unroll
        for (int t = 0; t < NHEAD; ++t)
            acc[t] += sc[wave][jj][t] * hp[32 * t];
    }
#pragma unroll
    for (int t = 0; t < NHEAD; ++t)
        out[(size_t)n * FOUT + 32 * t + lane] = acc[t];
}

// ---------------------------------------------------------------------------
extern "C" void kernel_launch(void* const* d_in, const int* in_sizes, int n_in,
                              void* d_out, int out_size, void* d_ws, size_t ws_size,
                              hipStream_t stream) {
    const float* x    = (const float*)d_in[0];  // (N, 512)
    const float* W    = (const float*)d_in[1];  // (8, 512, 32)
    const float* a    = (const float*)d_in[2];  // (8, 64)
    const int*   edge = (const int*)d_in[3];    // (2, E): row then col
    float* out = (float*)d_out;                 // (N, 256)

    // workspace: h (51.2MB) + s_src (1.6MB) + s_dst (1.6MB)
    float* h    = (float*)d_ws;
    float* ssrc = h + (size_t)NN * FOUT;
    float* sdst = ssrc + (size_t)NN * NHEAD;
    const int* colp = edge + (size_t)NN * DEG;  // second half of edge_index

    gat_gemm<<<(NN + BM - 1) / BM, 256, 0, stream>>>(x, W, h);
    gat_scores<<<(NN * NHEAD + 255) / 256, 256, 0, stream>>>(h, a, ssrc, sdst);
    gat_aggregate<<<NN / 8, 256, 0, stream>>>(h, ssrc, sdst, colp, out);
}